// VectorQuantizer_46943992545315
// MI455X (gfx1250) — compile-verified
//
#include <hip/hip_runtime.h>

typedef __attribute__((ext_vector_type(2))) float v2f;
typedef __attribute__((ext_vector_type(8))) float v8f;

#define BB 32768
#define DD 256
#define KK 1024

#define ROWS 64          // embedding rows per workgroup
#define TPB 64           // 2 waves; each wave owns 32 rows = two 16-row strips
#define SAD 260          // padded LDS row stride in floats (260 % 64 = 4 -> conflict-free)
#define NSTAGE 32        // codes staged per pass (two 16-wide WMMA N-tiles)
#define NPASS (KK / NSTAGE)

// ---------------------------------------------------------------------------
// Kernel 1: reciprocal squared norms of the codebook rows
// ---------------------------------------------------------------------------
__global__ void vq_norms_kernel(const float* __restrict__ cb, float* __restrict__ rnorm) {
  int k = blockIdx.x * blockDim.x + threadIdx.x;
  if (k < KK) {
    const float4* row = (const float4*)(cb + (size_t)k * DD);
    float s = 0.f;
#pragma unroll 8
    for (int i = 0; i < DD / 4; ++i) {
      float4 v = row[i];
      s += v.x * v.x + v.y * v.y + v.z * v.z + v.w * v.w;
    }
    rnorm[k] = 1.0f / s;
  }
}

// ---------------------------------------------------------------------------
// Kernel 2: fused GEMM (f32 WMMA, 2Mx2N register tile -> 4 accumulators)
//           + argmax(dot^2/|c|^2) + projection writeout
// ---------------------------------------------------------------------------
__launch_bounds__(TPB)
__global__ void vq_main_kernel(const float* __restrict__ emb,
                               const float* __restrict__ cb,
                               const float* __restrict__ rnorm,
                               float* __restrict__ zout,
                               float* __restrict__ idxout) {
  __shared__ float sA[ROWS * SAD];      // 64 x 256 embedding tile (padded)   ~66.5 KB
  __shared__ float sB[NSTAGE * SAD];    // 32 x 256 codebook chunk (padded)   ~33.3 KB

  const int tid  = threadIdx.x;
  const int wave = tid >> 5;
  const int lane = tid & 31;
  const int lr   = lane & 15;   // row-within-strip / col-within-tile selector
  const int half = lane >> 4;   // selects K pair {0,1} vs {2,3} in the 16x4 layout
  const int row0 = blockIdx.x * ROWS;
  const int m0   = wave * 32;   // this wave's 32-row block (strips m0 and m0+16)

  // Stage the 64x256 embedding tile into LDS (coalesced float4 loads).
  for (int i = tid; i < ROWS * (DD / 4); i += TPB) {
    int r = i >> 6, c = i & 63;
    ((float4*)(sA + r * SAD))[c] =
        ((const float4*)(emb + (size_t)(row0 + r) * DD))[c];
  }
  __syncthreads();

  // Per-strip running best (score, alpha, index), 8 row-slots per lane.
  float bs0[8], ba0[8], bs1[8], ba1[8];
  int   bn0[8], bn1[8];
#pragma unroll
  for (int v = 0; v < 8; ++v) {
    bs0[v] = -1.0f; ba0[v] = 0.0f; bn0[v] = 0;
    bs1[v] = -1.0f; ba1[v] = 0.0f; bn1[v] = 0;
  }

  // A operands: strip 0 and strip 1 rows
  const v2f* ap0 = (const v2f*)(sA + (m0 + lr) * SAD) + half;
  const v2f* ap1 = (const v2f*)(sA + (m0 + 16 + lr) * SAD) + half;

  for (int p = 0; p < NPASS; ++p) {
    const int n0 = p * NSTAGE;

    // Stage 32x256 codebook chunk into LDS.
    for (int i = tid; i < NSTAGE * (DD / 4); i += TPB) {
      int r = i >> 6, q = i & 63;
      ((float4*)(sB + r * SAD))[q] =
          ((const float4*)(cb + (size_t)(n0 + r) * DD))[q];
    }
    __syncthreads();

    // Prefetch next chunk (32 KB): four 128B lines per thread (64 threads).
    if (p + 1 < NPASS) {
      const float* nxt = cb + (size_t)(n0 + NSTAGE) * DD;
      __builtin_prefetch(nxt + tid * 32, 0, 0);
      __builtin_prefetch(nxt + 2048 + tid * 32, 0, 0);
      __builtin_prefetch(nxt + 4096 + tid * 32, 0, 0);
      __builtin_prefetch(nxt + 6144 + tid * 32, 0, 0);
    }

    v8f a00, a01, a10, a11;   // acc[Mstrip][Ntile]
#pragma unroll
    for (int v = 0; v < 8; ++v) { a00[v] = 0.f; a01[v] = 0.f; a10[v] = 0.f; a11[v] = 0.f; }

    // B operands: two 16-code tiles
    const v2f* bp0 = (const v2f*)(sB + lr * SAD) + half;
    const v2f* bp1 = (const v2f*)(sB + (16 + lr) * SAD) + half;

#pragma unroll
    for (int s = 0; s < DD / 4; ++s) {
      v2f A0 = ap0[2 * s];
      v2f A1 = ap1[2 * s];
      v2f B0 = bp0[2 * s];
      v2f B1 = bp1[2 * s];
      a00 = __builtin_amdgcn_wmma_f32_16x16x4_f32(false, A0, false, B0, (short)0, a00, false, false);
      a01 = __builtin_amdgcn_wmma_f32_16x16x4_f32(false, A0, false, B1, (short)0, a01, false, false);
      a10 = __builtin_amdgcn_wmma_f32_16x16x4_f32(false, A1, false, B0, (short)0, a10, false, false);
      a11 = __builtin_amdgcn_wmma_f32_16x16x4_f32(false, A1, false, B1, (short)0, a11, false, false);
    }

    const float rn0 = rnorm[n0 + lr];
    const float rn1 = rnorm[n0 + 16 + lr];
#pragma unroll
    for (int v = 0; v < 8; ++v) {
      // strip 0, tile 0 then tile 1 (ascending n keeps argmin-first tie-break)
      float d = a00[v]; float sc = d * d * rn0; bool bt = (sc > bs0[v]);
      bs0[v] = bt ? sc : bs0[v]; ba0[v] = bt ? d * rn0 : ba0[v]; bn0[v] = bt ? (n0 + lr) : bn0[v];
      d = a01[v]; sc = d * d * rn1; bt = (sc > bs0[v]);
      bs0[v] = bt ? sc : bs0[v]; ba0[v] = bt ? d * rn1 : ba0[v]; bn0[v] = bt ? (n0 + 16 + lr) : bn0[v];
      // strip 1
      d = a10[v]; sc = d * d * rn0; bt = (sc > bs1[v]);
      bs1[v] = bt ? sc : bs1[v]; ba1[v] = bt ? d * rn0 : ba1[v]; bn1[v] = bt ? (n0 + lr) : bn1[v];
      d = a11[v]; sc = d * d * rn1; bt = (sc > bs1[v]);
      bs1[v] = bt ? sc : bs1[v]; ba1[v] = bt ? d * rn1 : ba1[v]; bn1[v] = bt ? (n0 + 16 + lr) : bn1[v];
    }
    __syncthreads();   // protect sB before next pass's staging
  }

  // Butterfly-reduce the 16 candidate columns per row (stays within each half).
#pragma unroll
  for (int v = 0; v < 8; ++v) {
    float s0 = bs0[v], a0v = ba0[v]; int n0v = bn0[v];
    float s1 = bs1[v], a1v = ba1[v]; int n1v = bn1[v];
#pragma unroll
    for (int m = 1; m < 16; m <<= 1) {
      float so = __shfl_xor(s0, m, 32); float ao = __shfl_xor(a0v, m, 32); int no = __shfl_xor(n0v, m, 32);
      if (so > s0 || (so == s0 && no < n0v)) { s0 = so; a0v = ao; n0v = no; }
      so = __shfl_xor(s1, m, 32); ao = __shfl_xor(a1v, m, 32); no = __shfl_xor(n1v, m, 32);
      if (so > s1 || (so == s1 && no < n1v)) { s1 = so; a1v = ao; n1v = no; }
    }
    ba0[v] = a0v; bn0[v] = n0v;
    ba1[v] = a1v; bn1[v] = n1v;
  }

  // Writeout: 32 rows per wave; row (strip*16 + m) lives in slot (m&7) of half (m>>3).
#pragma unroll
  for (int m = 0; m < 32; ++m) {
    const int strip = m >> 4;
    const int mm = m & 15;
    const int v = mm & 7;
    const int h = mm >> 3;
    float alpha = strip ? __shfl(ba1[v], h * 16, 32) : __shfl(ba0[v], h * 16, 32);
    int   n     = strip ? __shfl(bn1[v], h * 16, 32) : __shfl(bn0[v], h * 16, 32);
    const int row = row0 + m0 + m;

    const float4* crow = (const float4*)(cb + (size_t)n * DD);
    float4* zrow = (float4*)(zout + (size_t)row * DD);
    float4 c0 = crow[lane];
    float4 c1 = crow[lane + 32];
    zrow[lane]      = make_float4(alpha * c0.x, alpha * c0.y, alpha * c0.z, alpha * c0.w);
    zrow[lane + 32] = make_float4(alpha * c1.x, alpha * c1.y, alpha * c1.z, alpha * c1.w);
    if (lane == 0) idxout[row] = (float)n;
  }
}

// ---------------------------------------------------------------------------
extern "C" void kernel_launch(void* const* d_in, const int* in_sizes, int n_in,
                              void* d_out, int out_size, void* d_ws, size_t ws_size,
                              hipStream_t stream) {
  const float* emb = (const float*)d_in[0];   // (B, D) fp32
  const float* cb  = (const float*)d_in[1];   // (K, D) fp32

  float* zout   = (float*)d_out;                      // (B, D)
  float* idxout = (float*)d_out + (size_t)BB * DD;    // (B,) indices as float
  float* rnorm  = (float*)d_ws;                       // (K,) scratch

  vq_norms_kernel<<<KK / 256, 256, 0, stream>>>(cb, rnorm);
  vq_main_kernel<<<BB / ROWS, TPB, 0, stream>>>(emb, cb, rnorm, zout, idxout);
}